// GAT_GResNet_4011499454862
// MI455X (gfx1250) — compile-verified
//
#include <hip/hip_runtime.h>
#include <math.h>

// ---- problem constants (from reference) ----
#define N_NODES 50000
#define N_EDGES 800000
#define IN_DIM  300
#define HID     16
#define HEADS   8
#define OUT_DIM 10
#define HW      128      // HID*HEADS
#define NEG_SLOPE 0.2f

typedef float v2f __attribute__((ext_vector_type(2)));
typedef float v8f __attribute__((ext_vector_type(8)));

static inline int cdiv(long a, long b) { return (int)((a + b - 1) / b); }

// ---------------------------------------------------------------------------
// f32 WMMA GEMM: C[N x M] = A[N x K] @ W[K x M], row-major.
// Each wave computes a 16 x (16*NT) strip of C. A fragment loaded once per
// K-step and reused across NT B column-tiles (NT back-to-back WMMAs).
// M is a template constant so B-row strides fold into immediate offsets.
// Ragged M: out-of-range columns read clamped (valid, finite) W column 0;
// by WMMA semantics a B column only affects the same output column, which we
// never store -> no masking needed, no divergence around the WMMA.
// Software-pipelined: next K-step fragments are loaded before current WMMAs.
// Requires: N % 16 == 0, K % 4 == 0, K >= 8.
// ---------------------------------------------------------------------------
template <int NT, int M>
__global__ void wmma_gemm_f32_nt(const float* __restrict__ A,
                                 const float* __restrict__ W,
                                 float* __restrict__ C,
                                 int K, int MtileGroups, int tiles_total) {
  int wave = blockIdx.x * (blockDim.x >> 5) + (threadIdx.x >> 5);
  if (wave >= tiles_total) return;   // whole-wave exit: EXEC stays all-ones
  int lane = threadIdx.x & 31;
  int tg = wave % MtileGroups;       // column tile-group
  int tm = wave / MtileGroups;       // row tile
  int m0 = tm * 16;
  int r15  = lane & 15;
  int kgrp = (lane >> 4) * 2;        // lanes 0-15 -> k+0/k+1 ; lanes 16-31 -> k+2/k+3

  const float* aptr = A + (size_t)(m0 + r15) * K + kgrp;

  const float* bptr[NT];
  int bcol[NT];
#pragma unroll
  for (int t = 0; t < NT; ++t) {
    int col = (tg * NT + t) * 16 + r15;
    bcol[t] = col;
    bptr[t] = W + (size_t)kgrp * M + (col < M ? col : 0);
  }

  v8f acc[NT];
#pragma unroll
  for (int t = 0; t < NT; ++t) acc[t] = (v8f){};

  // prologue: fragments for k = 0
  v2f a, b[NT];
  a.x = aptr[0];
  a.y = aptr[1];
#pragma unroll
  for (int t = 0; t < NT; ++t) { b[t].x = bptr[t][0]; b[t].y = bptr[t][M]; }

  for (int k = 0; k < K - 4; k += 4) {
    aptr += 4;
#pragma unroll
    for (int t = 0; t < NT; ++t) bptr[t] += 4 * M;
    // prefetch next K-step fragments (loads overlap the WMMAs below)
    v2f an;
    an.x = aptr[0];
    an.y = aptr[1];
    v2f bn[NT];
#pragma unroll
    for (int t = 0; t < NT; ++t) { bn[t].x = bptr[t][0]; bn[t].y = bptr[t][M]; }
    // (neg_a, A, neg_b, B, c_mod, C, reuse_a, reuse_b)
#pragma unroll
    for (int t = 0; t < NT; ++t)
      acc[t] = __builtin_amdgcn_wmma_f32_16x16x4_f32(false, a, false, b[t],
                                                     (short)0, acc[t], false, false);
    a = an;
#pragma unroll
    for (int t = 0; t < NT; ++t) b[t] = bn[t];
  }
  // epilogue
#pragma unroll
  for (int t = 0; t < NT; ++t)
    acc[t] = __builtin_amdgcn_wmma_f32_16x16x4_f32(false, a, false, b[t],
                                                   (short)0, acc[t], false, false);

  int rbase = m0 + ((lane >> 4) * 8);   // lanes 16-31 hold rows m0+8..m0+15
#pragma unroll
  for (int t = 0; t < NT; ++t) {
    if (bcol[t] < M) {
#pragma unroll
      for (int r = 0; r < 8; ++r)
        C[(size_t)(rbase + r) * M + bcol[t]] = acc[t][r];
    }
  }
}

// ---------------------------------------------------------------------------
// graph / elementwise kernels
// ---------------------------------------------------------------------------
__global__ void deg_kernel(const int* __restrict__ dst, float* __restrict__ deg) {
  int e = blockIdx.x * blockDim.x + threadIdx.x;
  if (e < N_EDGES) atomicAdd(&deg[dst[e]], 1.0f);
}

__global__ void norm_kernel(const float* __restrict__ deg, float* __restrict__ nrm) {
  int n = blockIdx.x * blockDim.x + threadIdx.x;
  if (n < N_NODES) nrm[n] = rsqrtf(fmaxf(deg[n], 1.0f));
}

// el[n,h] = sum_d feat[n,h,d]*al[h,d] ; er likewise
__global__ void attn_kernel(const float* __restrict__ feat,
                            const float* __restrict__ al,
                            const float* __restrict__ ar,
                            float* __restrict__ el, float* __restrict__ er,
                            int H, int D) {
  int i = blockIdx.x * blockDim.x + threadIdx.x;
  if (i >= N_NODES * H) return;
  int n = i / H, h = i % H;
  const float* f = feat + (size_t)n * H * D + (size_t)h * D;
  float sl = 0.f, sr = 0.f;
  for (int d = 0; d < D; ++d) { sl += f[d] * al[h * D + d]; sr += f[d] * ar[h * D + d]; }
  el[i] = sl; er[i] = sr;
}

// order-preserving float<->uint for atomicMax on floats
__device__ __forceinline__ unsigned f2key(float f) {
  unsigned b = __float_as_uint(f);
  return (b & 0x80000000u) ? ~b : (b | 0x80000000u);
}
__device__ __forceinline__ float key2f(unsigned k) {
  unsigned b = (k & 0x80000000u) ? (k & 0x7FFFFFFFu) : ~k;
  return __uint_as_float(b);
}

__device__ __forceinline__ float leaky(float x) {
  return x > 0.0f ? x : NEG_SLOPE * x;
}

__global__ void edge_max_kernel(const int* __restrict__ src, const int* __restrict__ dst,
                                const float* __restrict__ el, const float* __restrict__ er,
                                unsigned* __restrict__ key, int H) {
  int i = blockIdx.x * blockDim.x + threadIdx.x;
  if (i >= N_EDGES * H) return;
  int e = i / H, h = i % H;
  int s = src[e], d = dst[e];
  float lr = leaky(el[s * H + h] + er[d * H + h]);
  atomicMax(&key[d * H + h], f2key(lr));
}

__global__ void decode_max_kernel(float* __restrict__ emax, int count) {
  int i = blockIdx.x * blockDim.x + threadIdx.x;
  if (i < count) {
    unsigned k = ((unsigned*)emax)[i];
    emax[i] = key2f(k);
  }
}

__global__ void edge_expsum_kernel(const int* __restrict__ src, const int* __restrict__ dst,
                                   const float* __restrict__ el, const float* __restrict__ er,
                                   const float* __restrict__ emax, float* __restrict__ denom,
                                   int H) {
  int i = blockIdx.x * blockDim.x + threadIdx.x;
  if (i >= N_EDGES * H) return;
  int e = i / H, h = i % H;
  int s = src[e], d = dst[e];
  float lr = leaky(el[s * H + h] + er[d * H + h]);
  atomicAdd(&denom[d * H + h], __expf(lr - emax[d * H + h]));
}

// out[dst, h, :] += alpha * feat[src, h, :]
__global__ void edge_aggr_kernel(const int* __restrict__ src, const int* __restrict__ dst,
                                 const float* __restrict__ el, const float* __restrict__ er,
                                 const float* __restrict__ emax, const float* __restrict__ denom,
                                 const float* __restrict__ feat, float* __restrict__ out,
                                 int H, int D) {
  int i = blockIdx.x * blockDim.x + threadIdx.x;
  if (i >= N_EDGES * H) return;
  int e = i / H, h = i % H;
  int s = src[e], d = dst[e];
  float lr = leaky(el[s * H + h] + er[d * H + h]);
  float alpha = __expf(lr - emax[d * H + h]) / denom[d * H + h];
  const float* fs = feat + (size_t)s * H * D + (size_t)h * D;
  float*       od = out  + (size_t)d * H * D + (size_t)h * D;
  for (int dd = 0; dd < D; ++dd) atomicAdd(&od[dd], alpha * fs[dd]);
}

// acc[dst, f] += res[src, f] * norm[src]
__global__ void res_edge_kernel(const int* __restrict__ src, const int* __restrict__ dst,
                                const float* __restrict__ res, const float* __restrict__ nrm,
                                float* __restrict__ acc, int F) {
  int i = blockIdx.x * blockDim.x + threadIdx.x;
  if (i >= N_EDGES * F) return;
  int e = i / F, f = i % F;
  int s = src[e], d = dst[e];
  atomicAdd(&acc[(size_t)d * F + f], res[(size_t)s * F + f] * nrm[s]);
}

__global__ void scale_rows_kernel(const float* __restrict__ acc, const float* __restrict__ nrm,
                                  float* __restrict__ out, int F) {
  int i = blockIdx.x * blockDim.x + threadIdx.x;
  if (i >= N_NODES * F) return;
  out[i] = acc[i] * nrm[i / F];
}

__global__ void add_elu_kernel(const float* __restrict__ a, const float* __restrict__ b,
                               float* __restrict__ h, int count) {
  int i = blockIdx.x * blockDim.x + threadIdx.x;
  if (i >= count) return;
  float x = a[i] + b[i];
  h[i] = x > 0.0f ? x : (__expf(x) - 1.0f);
}

// out[n,f] += acc[n,f]*norm[n]
__global__ void final_add_kernel(float* __restrict__ out, const float* __restrict__ acc,
                                 const float* __restrict__ nrm) {
  int i = blockIdx.x * blockDim.x + threadIdx.x;
  if (i >= N_NODES * OUT_DIM) return;
  out[i] += acc[i] * nrm[i / OUT_DIM];
}

// ---------------------------------------------------------------------------
extern "C" void kernel_launch(void* const* d_in, const int* in_sizes, int n_in,
                              void* d_out, int out_size, void* d_ws, size_t ws_size,
                              hipStream_t stream) {
  const float* features = (const float*)d_in[0];
  const int*   src      = (const int*)d_in[1];
  const int*   dst      = (const int*)d_in[2];
  const float* W_raw_in  = (const float*)d_in[3];
  const float* W_raw_out = (const float*)d_in[4];
  const float* W1  = (const float*)d_in[5];
  const float* al1 = (const float*)d_in[6];
  const float* ar1 = (const float*)d_in[7];
  const float* W2  = (const float*)d_in[8];
  const float* al2 = (const float*)d_in[9];
  const float* ar2 = (const float*)d_in[10];
  const float* Wo  = (const float*)d_in[11];
  const float* alo = (const float*)d_in[12];
  const float* aro = (const float*)d_in[13];
  float* out = (float*)d_out;

  const size_t N = N_NODES;
  float* ws    = (float*)d_ws;
  float* deg   = ws;                    // N
  float* nrm   = deg + N;               // N
  float* raw   = nrm + N;               // N*HW
  float* res_h = raw + N * HW;          // N*HW
  float* feat  = res_h + N * HW;        // N*HW (reused for 10-wide output feats)
  float* hbuf  = feat + N * HW;         // N*HW
  float* acc   = hbuf + N * HW;         // N*HW (generic accumulator)
  float* el    = acc + N * HW;          // N*HEADS
  float* er    = el + N * HEADS;        // N*HEADS
  float* emax  = er + N * HEADS;        // N*HEADS (also used as uint keys)
  float* denom = emax + N * HEADS;      // N*HEADS
  float* raw_o = denom + N * HEADS;     // N*OUT_DIM
  float* acco  = raw_o + N * OUT_DIM;   // N*OUT_DIM

  const int B = 256;
  const int NTILES = N_NODES / 16;          // 3125 (exact)
  const int WPB = B / 32;                   // waves per block

  // wide GEMMs (M=128): 4 column tiles per wave -> 2 tile-groups
  const int GROUPS_W = (HW / 16) / 4;       // 2
  const int TILES_W  = NTILES * GROUPS_W;   // 6250 waves
  // narrow GEMMs (M=10): 1 column tile per wave
  const int TILES_N  = NTILES;              // 3125 waves

  // degree + norm
  hipMemsetAsync(deg, 0, N * sizeof(float), stream);
  deg_kernel<<<cdiv(N_EDGES, B), B, 0, stream>>>(dst, deg);
  norm_kernel<<<cdiv(N_NODES, B), B, 0, stream>>>(deg, nrm);

  // raw = features @ W_raw_in  [N,300]x[300,128]
  wmma_gemm_f32_nt<4, HW><<<cdiv(TILES_W, WPB), B, 0, stream>>>(features, W_raw_in, raw,
                                                                IN_DIM, GROUPS_W, TILES_W);
  // res_h = graph_res(raw)
  hipMemsetAsync(acc, 0, N * HW * sizeof(float), stream);
  res_edge_kernel<<<cdiv((long)N_EDGES * HW, B), B, 0, stream>>>(src, dst, raw, nrm, acc, HW);
  scale_rows_kernel<<<cdiv(N * HW, B), B, 0, stream>>>(acc, nrm, res_h, HW);

  // layer 1: feat = features @ W1
  wmma_gemm_f32_nt<4, HW><<<cdiv(TILES_W, WPB), B, 0, stream>>>(features, W1, feat,
                                                                IN_DIM, GROUPS_W, TILES_W);
  attn_kernel<<<cdiv(N * HEADS, B), B, 0, stream>>>(feat, al1, ar1, el, er, HEADS, HID);
  hipMemsetAsync(emax, 0, N * HEADS * sizeof(float), stream);
  edge_max_kernel<<<cdiv((long)N_EDGES * HEADS, B), B, 0, stream>>>(src, dst, el, er,
                                                                    (unsigned*)emax, HEADS);
  decode_max_kernel<<<cdiv(N * HEADS, B), B, 0, stream>>>(emax, (int)(N * HEADS));
  hipMemsetAsync(denom, 0, N * HEADS * sizeof(float), stream);
  edge_expsum_kernel<<<cdiv((long)N_EDGES * HEADS, B), B, 0, stream>>>(src, dst, el, er,
                                                                      emax, denom, HEADS);
  hipMemsetAsync(acc, 0, N * HW * sizeof(float), stream);
  edge_aggr_kernel<<<cdiv((long)N_EDGES * HEADS, B), B, 0, stream>>>(src, dst, el, er, emax,
                                                                    denom, feat, acc, HEADS, HID);
  add_elu_kernel<<<cdiv(N * HW, B), B, 0, stream>>>(acc, res_h, hbuf, (int)(N * HW));

  // layer 2: feat = h @ W2
  wmma_gemm_f32_nt<4, HW><<<cdiv(TILES_W, WPB), B, 0, stream>>>(hbuf, W2, feat,
                                                                HW, GROUPS_W, TILES_W);
  attn_kernel<<<cdiv(N * HEADS, B), B, 0, stream>>>(feat, al2, ar2, el, er, HEADS, HID);
  hipMemsetAsync(emax, 0, N * HEADS * sizeof(float), stream);
  edge_max_kernel<<<cdiv((long)N_EDGES * HEADS, B), B, 0, stream>>>(src, dst, el, er,
                                                                    (unsigned*)emax, HEADS);
  decode_max_kernel<<<cdiv(N * HEADS, B), B, 0, stream>>>(emax, (int)(N * HEADS));
  hipMemsetAsync(denom, 0, N * HEADS * sizeof(float), stream);
  edge_expsum_kernel<<<cdiv((long)N_EDGES * HEADS, B), B, 0, stream>>>(src, dst, el, er,
                                                                      emax, denom, HEADS);
  hipMemsetAsync(acc, 0, N * HW * sizeof(float), stream);
  edge_aggr_kernel<<<cdiv((long)N_EDGES * HEADS, B), B, 0, stream>>>(src, dst, el, er, emax,
                                                                    denom, feat, acc, HEADS, HID);
  add_elu_kernel<<<cdiv(N * HW, B), B, 0, stream>>>(acc, res_h, hbuf, (int)(N * HW));

  // output stage
  // raw_o = raw @ W_raw_out  [N,128]x[128,10]
  wmma_gemm_f32_nt<1, OUT_DIM><<<cdiv(TILES_N, WPB), B, 0, stream>>>(raw, W_raw_out, raw_o,
                                                                    HW, 1, TILES_N);
  // feato = h @ Wo (stored in feat buffer, N x 10)
  wmma_gemm_f32_nt<1, OUT_DIM><<<cdiv(TILES_N, WPB), B, 0, stream>>>(hbuf, Wo, feat,
                                                                    HW, 1, TILES_N);
  attn_kernel<<<cdiv(N * 1, B), B, 0, stream>>>(feat, alo, aro, el, er, 1, OUT_DIM);
  hipMemsetAsync(emax, 0, N * sizeof(float), stream);
  edge_max_kernel<<<cdiv((long)N_EDGES, B), B, 0, stream>>>(src, dst, el, er,
                                                            (unsigned*)emax, 1);
  decode_max_kernel<<<cdiv(N, B), B, 0, stream>>>(emax, (int)N);
  hipMemsetAsync(denom, 0, N * sizeof(float), stream);
  edge_expsum_kernel<<<cdiv((long)N_EDGES, B), B, 0, stream>>>(src, dst, el, er,
                                                               emax, denom, 1);
  hipMemsetAsync(out, 0, (size_t)out_size * sizeof(float), stream);
  edge_aggr_kernel<<<cdiv((long)N_EDGES, B), B, 0, stream>>>(src, dst, el, er, emax,
                                                             denom, feat, out, 1, OUT_DIM);
  // + graph_res(raw_o)
  hipMemsetAsync(acco, 0, N * OUT_DIM * sizeof(float), stream);
  res_edge_kernel<<<cdiv((long)N_EDGES * OUT_DIM, B), B, 0, stream>>>(src, dst, raw_o, nrm,
                                                                     acco, OUT_DIM);
  final_add_kernel<<<cdiv(N * OUT_DIM, B), B, 0, stream>>>(out, acco, nrm);
}